// PointCloudGrouper_40630390620888
// MI455X (gfx1250) — compile-verified
//
#include <hip/hip_runtime.h>
#include <hip/hip_bf16.h>

typedef __attribute__((ext_vector_type(2))) float v2f;
typedef __attribute__((ext_vector_type(8))) float v8f;

#define NPTS      8192
#define NCENT     27
#define NSAMPLE   512
#define CHUNK     512          // points per LDS chunk
#define NCHUNKS   (NPTS / CHUNK)
#define R2        0.0625f      // RADIUS^2
#define NWAVES    27
#define BLOCK     (NWAVES * 32)
// LDS layout (floats): xyz chunk [CHUNK*3] then dist2 [32*CHUNK]
#define XYZC_F    (CHUNK * 3)
#define DIST_F    (32 * CHUNK)
#define SMEM_BYTES ((XYZC_F + DIST_F) * sizeof(float))
// one B128 async transfer = 16 bytes; chunk = CHUNK*3*4 bytes
#define ASYNC_LANES ((CHUNK * 3 * 4) / 16)   // 384

#if defined(__has_builtin)
#if __has_builtin(__builtin_amdgcn_global_load_async_to_lds_b128)
#define HAVE_ASYNC_LDS 1
#endif
#if __has_builtin(__builtin_amdgcn_s_wait_asynccnt)
#define HAVE_WAIT_ASYNC 1
#endif
#endif

// builtin payload type per compiler diagnostic: v4i32 pointers in AS1 / AS3
typedef int v4i_vs __attribute__((__vector_size__(4 * sizeof(int))));
typedef __attribute__((address_space(1))) v4i_vs* as1_v4i;
typedef __attribute__((address_space(3))) v4i_vs* as3_v4i;

__global__ __launch_bounds__(BLOCK)
void PointCloudGrouper_40630390620888_kernel(const float* __restrict__ xyz,
                                             const float* __restrict__ centers,
                                             float* __restrict__ out) {
    extern __shared__ float smem[];
    float* xyzc  = smem;             // [CHUNK][3]
    float* dist2 = smem + XYZC_F;    // [32][CHUNK]

    const int b    = blockIdx.x;
    const int tid  = threadIdx.x;
    const int w    = tid >> 5;       // wave id == center id (0..26)
    const int lane = tid & 31;
    const int lo16 = lane & 15;
    const bool hi  = lane >= 16;

    const float* xb = xyz + (size_t)b * (NPTS * 3);

    // ---- Precompute A fragments (chunk-invariant): V_WMMA_F32_16X16X4_F32 A layout ----
    // lane l: M = l%16 ; K = (l<16 ? 0 : 2) + r.  A row p = [-2cx, -2cy, -2cz, |c|^2]
    v2f aFrag[2];
#pragma unroll
    for (int pt = 0; pt < 2; ++pt) {
        int M = pt * 16 + lo16;
        float v0 = 0.0f, v1 = 0.0f;
        if (M < NCENT) {
            float ccx = centers[M * 3 + 0];
            float ccy = centers[M * 3 + 1];
            float ccz = centers[M * 3 + 2];
            float c2  = ccx * ccx + ccy * ccy + ccz * ccz;
            if (!hi) { v0 = -2.0f * ccx; v1 = -2.0f * ccy; }
            else     { v0 = -2.0f * ccz; v1 = c2; }
        }
        aFrag[pt][0] = v0;
        aFrag[pt][1] = v1;
    }

    // This wave's center (for output re-centering)
    const float cx = centers[w * 3 + 0];
    const float cy = centers[w * 3 + 1];
    const float cz = centers[w * 3 + 2];

    int  base     = 0;   // points found so far for (b, w)
    int  firstIdx = -1;  // first in-radius global point index

    for (int chunk = 0; chunk < NCHUNKS; ++chunk) {
        // ---- Stage xyz chunk to LDS: async DMA (ASYNCcnt path) when available ----
        const float* src = xb + (size_t)chunk * (CHUNK * 3);
#if HAVE_ASYNC_LDS
        if (tid < ASYNC_LANES) {
            const char* sp = (const char*)src + tid * 16;
            char*       lp = (char*)xyzc + tid * 16;
            __builtin_amdgcn_global_load_async_to_lds_b128(
                (as1_v4i)(void*)sp, (as3_v4i)(void*)lp, /*offset=*/0, /*cpol=*/0);
        }
#if HAVE_WAIT_ASYNC
        __builtin_amdgcn_s_wait_asynccnt(0);
#else
        asm volatile("s_wait_asynccnt 0" ::: "memory");
#endif
#else
        for (int i = tid; i < CHUNK * 3; i += BLOCK)
            xyzc[i] = src[i];
#endif
        if (chunk + 1 < NCHUNKS) {
            const float* nxt = xb + (size_t)(chunk + 1) * (CHUNK * 3);
            // one prefetch per wave, spread across the chunk's cachelines
            __builtin_prefetch(nxt + w * 56, 0, 0);   // -> global_prefetch_b8
        }
        __syncthreads();

        // ---- Phase 1: WMMA dist^2 tiles (2 p-tiles x 32 n-tiles = 64 tiles) ----
        for (int t = w; t < 64; t += NWAVES) {        // wave-uniform loop: EXEC stays all-1s
            const int pt = t >> 5;
            const int nt = t & 31;
            const int nl = nt * 16 + lo16;            // local point this lane touches
            const float bx = xyzc[nl * 3 + 0];
            const float by = xyzc[nl * 3 + 1];
            const float bz = xyzc[nl * 3 + 2];
            v2f bFrag;
            if (!hi) { bFrag[0] = bx; bFrag[1] = by; }     // K=0,1
            else     { bFrag[0] = bz; bFrag[1] = 1.0f; }   // K=2,3 (augmented 1)
            v8f acc = {};
            acc = __builtin_amdgcn_wmma_f32_16x16x4_f32(
                /*neg_a=*/false, aFrag[pt], /*neg_b=*/false, bFrag,
                /*c_mod=*/(short)0, acc, /*reuse_a=*/false, /*reuse_b=*/false);
            const float x2 = bx * bx + by * by + bz * bz;  // lane's column N fixed -> add |x|^2
            const int Mb = hi ? 8 : 0;
#pragma unroll
            for (int r = 0; r < 8; ++r)
                dist2[(pt * 16 + Mb + r) * CHUNK + nl] = acc[r] + x2;
        }
        __syncthreads();

        // ---- Phase 2: ordered ball-query compaction, wave w scans row w ----
#pragma unroll 4
        for (int i = 0; i < CHUNK / 32; ++i) {
            const int jl = i * 32 + lane;                       // local point index
            const float d2 = dist2[w * CHUNK + jl];
            const bool  m  = d2 < R2;
            const unsigned bal = __builtin_amdgcn_ballot_w32(m);
            if (firstIdx < 0 && bal != 0u)
                firstIdx = chunk * CHUNK + i * 32 + __builtin_ctz(bal);
            const unsigned pre = __popc(bal & ((1u << lane) - 1u));
            const int slot = base + (int)pre;
            if (m && slot < NSAMPLE) {
                float* o = out + (((size_t)(b * NCENT + w)) * NSAMPLE + slot) * 3;
                o[0] = xyzc[jl * 3 + 0] - cx;
                o[1] = xyzc[jl * 3 + 1] - cy;
                o[2] = xyzc[jl * 3 + 2] - cz;
            }
            base += __popc(bal);
        }
        __syncthreads();   // dist2 / xyzc reused next chunk
    }

    // ---- Tail padding: slots [base, NSAMPLE) get first found point (0 if none) ----
    const int fi = (firstIdx < 0) ? 0 : firstIdx;
    const float fx = xb[(size_t)fi * 3 + 0] - cx;
    const float fy = xb[(size_t)fi * 3 + 1] - cy;
    const float fz = xb[(size_t)fi * 3 + 2] - cz;
    for (int s = base + lane; s < NSAMPLE; s += 32) {
        float* o = out + (((size_t)(b * NCENT + w)) * NSAMPLE + s) * 3;
        o[0] = fx; o[1] = fy; o[2] = fz;
    }
}

extern "C" void kernel_launch(void* const* d_in, const int* in_sizes, int n_in,
                              void* d_out, int out_size, void* d_ws, size_t ws_size,
                              hipStream_t stream) {
    (void)in_sizes; (void)n_in; (void)d_ws; (void)ws_size; (void)out_size;
    const float* xyz     = (const float*)d_in[0];   // [128, 8192, 3]
    const float* centers = (const float*)d_in[1];   // [1, 27, 3]
    float* out = (float*)d_out;                     // [128*27, 512, 3]

    dim3 grid(128);
    dim3 block(BLOCK);
    PointCloudGrouper_40630390620888_kernel<<<grid, block, SMEM_BYTES, stream>>>(xyz, centers, out);
}